// GNNModule_15470472200655
// MI455X (gfx1250) — compile-verified
//
#include <hip/hip_runtime.h>

typedef float v2f __attribute__((ext_vector_type(2)));
typedef float v8f __attribute__((ext_vector_type(8)));

#define BS 256

// ---------------- utility fill ----------------
__global__ void k_fill(float* __restrict__ p, int n, float v) {
    int i = blockIdx.x * blockDim.x + threadIdx.x;
    if (i < n) p[i] = v;
}

// ---------------- degree (deg pre-set to 1.0 = self-loop) ----------------
__global__ void k_degree(const int* __restrict__ ei, float* __restrict__ deg, int E) {
    int e = blockIdx.x * blockDim.x + threadIdx.x;
    if (e < E) unsafeAtomicAdd(&deg[ei[E + e]], 1.0f);
}

// deg >= 1 always, so rsqrt matches the reference's where(deg>0) guard.
__global__ void k_rsqrt(float* __restrict__ d, int N) {
    int i = blockIdx.x * blockDim.x + threadIdx.x;
    if (i < N) d[i] = rsqrtf(d[i]);
}

// ---------------- layer-1 aggregation on RAW x (3-wide) + norm cache ----------------
// One thread per edge (incl. self-loops at [E, E+N)). Computes norm once, stores it
// for the layer-2 edge kernel, and scatter-adds x[src]*norm into agg3[dst].
__global__ void k_edge3(const int* __restrict__ ei, const float* __restrict__ dinv,
                        const float* __restrict__ x, float* __restrict__ agg3,
                        float* __restrict__ nrm_out, int E, int Nn) {
    int e = blockIdx.x * blockDim.x + threadIdx.x;
    if (e >= E + Nn) return;
    int s, d;
    if (e < E) { s = ei[e]; d = ei[E + e]; }
    else       { s = e - E; d = s; }
    float nrm = dinv[s] * dinv[d];
    nrm_out[e] = nrm;
    float* a = &agg3[(size_t)d * 3];
    unsafeAtomicAdd(a + 0, x[(size_t)s * 3 + 0] * nrm);
    unsafeAtomicAdd(a + 1, x[(size_t)s * 3 + 1] * nrm);
    unsafeAtomicAdd(a + 2, x[(size_t)s * 3 + 2] * nrm);
}

// ---------------- h1 = relu(agg3 @ W1 + b1) via WMMA f32 16x16x4 (K padded 3->4) ----
// Block: 8 waves = 2 row-tiles x 4 col-tiles (64 cols). Grid covers N in 32-row chunks.
__global__ void k_gemm1(const float* __restrict__ agg3, const float* __restrict__ W1,
                        const float* __restrict__ b1, float* __restrict__ h1, int N) {
    int lane = threadIdx.x & 31;
    int wave = threadIdx.x >> 5;
    int n0 = (wave & 3) * 16;                     // col tile
    int m0 = (blockIdx.x * 2 + (wave >> 2)) * 16; // row tile
    int row = lane & 15;
    bool hi = (lane >> 4) != 0;

    int mrow = m0 + row;
    if (mrow >= N) mrow = N - 1;                  // keep EXEC all-ones
    const float* ar = agg3 + (size_t)mrow * 3;

    // Unconditional loads (always in-bounds), branch-free v_cndmask selects.
    float f0 = ar[0], f1 = ar[1], f2 = ar[2];
    float w0 = W1[0 * 64 + n0 + row];
    float w1 = W1[1 * 64 + n0 + row];
    float w2 = W1[2 * 64 + n0 + row];

    // A 16x4: lanes 0-15 K={0,1}; lanes 16-31 K={2,3} with K3 = 0 (pad)
    v2f a, b;
    a.x = hi ? f2 : f0;
    a.y = hi ? 0.f : f1;
    b.x = hi ? w2 : w0;
    b.y = hi ? 0.f : w1;

    v8f c = {};
    c = __builtin_amdgcn_wmma_f32_16x16x4_f32(false, a, false, b, (short)0, c, false, false);

    float bias = b1[n0 + row];
    int rbase = m0 + (hi ? 8 : 0);
    float* hb = h1 + (size_t)rbase * 64 + n0 + row;
    if (m0 + 16 <= N) {                           // wave-uniform: full tile, no guards
#pragma unroll
        for (int v = 0; v < 8; ++v) {
            float val = c[v] + bias;
            hb[(size_t)v * 64] = val > 0.f ? val : 0.f;
        }
    } else {                                      // tail tile only
#pragma unroll
        for (int v = 0; v < 8; ++v) {
            if (rbase + v < N) {
                float val = c[v] + bias;
                hb[(size_t)v * 64] = val > 0.f ? val : 0.f;
            }
        }
    }
}

// ---------------- layer-2 aggregation: agg64[dst] += h1[src]*norm (dominant kernel) --
// 16 threads per edge, float4 per thread; norm read is a per-edge broadcast.
__global__ void k_edge64(const int* __restrict__ ei, const float* __restrict__ nrm,
                         const float* __restrict__ h1, float* __restrict__ agg,
                         int E, int Nn) {
    int t = blockIdx.x * blockDim.x + threadIdx.x;
    if (t >= (E + Nn) * 16) return;
    int e = t >> 4;
    int f = (t & 15) * 4;
    int s, d;
    if (e < E) { s = ei[e]; d = ei[E + e]; }
    else       { s = e - E; d = s; }
    float w = nrm[e];
    const float4 v = *(const float4*)&h1[(size_t)s * 64 + f];
    float* a = &agg[(size_t)d * 64 + f];
    unsafeAtomicAdd(a + 0, v.x * w);
    unsafeAtomicAdd(a + 1, v.y * w);
    unsafeAtomicAdd(a + 2, v.z * w);
    unsafeAtomicAdd(a + 3, v.w * w);
}

// ---------------- graph pooling of agg64 -> pooled[128,64], counts[128] -------------
__global__ void k_pool(const float* __restrict__ agg64, const int* __restrict__ batch,
                       float* __restrict__ pooled, float* __restrict__ counts, int N) {
    int t = blockIdx.x * blockDim.x + threadIdx.x;
    if (t >= N * 16) return;
    int node = t >> 4;
    int c0 = (t & 15) * 4;
    int g = batch[node];
    const float4 v = *(const float4*)&agg64[(size_t)node * 64 + c0];
    float* o = &pooled[(size_t)g * 64 + c0];
    unsafeAtomicAdd(o + 0, v.x);
    unsafeAtomicAdd(o + 1, v.y);
    unsafeAtomicAdd(o + 2, v.z);
    unsafeAtomicAdd(o + 3, v.w);
    if (c0 == 0) unsafeAtomicAdd(&counts[g], 1.0f);
}

// pooled /= max(counts, 1)   (mean before the final linear map)
__global__ void k_prep(float* __restrict__ pooled, const float* __restrict__ counts) {
    int i = blockIdx.x * blockDim.x + threadIdx.x;
    if (i < 128 * 64) pooled[i] /= fmaxf(counts[i >> 6], 1.0f);
}

// ---------------- out[128,128] = pooled @ W2 + (counts>0 ? b2 : 0) via WMMA ---------
// Grid: 8 blocks (row tiles) x 8 waves (col tiles); K=64 -> 16 WMMA per wave.
__global__ void k_gemm_out(const float* __restrict__ pooled, const float* __restrict__ W2,
                           const float* __restrict__ b2, const float* __restrict__ counts,
                           float* __restrict__ out) {
    int lane = threadIdx.x & 31;
    int wave = threadIdx.x >> 5;
    int m0 = blockIdx.x * 16;
    int n0 = wave * 16;
    int row = lane & 15;
    int hi  = lane >> 4;

    const float* ar = pooled + (size_t)(m0 + row) * 64;

    v8f c = {};
#pragma unroll
    for (int k = 0; k < 64; k += 4) {
        v2f a, b;
        a.x = ar[k + 2 * hi + 0];
        a.y = ar[k + 2 * hi + 1];
        b.x = W2[(size_t)(k + 2 * hi + 0) * 128 + n0 + row];
        b.y = W2[(size_t)(k + 2 * hi + 1) * 128 + n0 + row];
        c = __builtin_amdgcn_wmma_f32_16x16x4_f32(false, a, false, b, (short)0, c, false, false);
    }

    float bias = b2[n0 + row];
#pragma unroll
    for (int v = 0; v < 8; ++v) {
        int r = m0 + v + 8 * hi;                 // graph index
        float sel = counts[r] > 0.f ? 1.f : 0.f; // empty graphs -> exact 0 like reference
        out[(size_t)r * 128 + n0 + row] = c[v] + sel * bias;
    }
}

// ---------------- host launcher ----------------
extern "C" void kernel_launch(void* const* d_in, const int* in_sizes, int n_in,
                              void* d_out, int out_size, void* d_ws, size_t ws_size,
                              hipStream_t stream) {
    const float* x     = (const float*)d_in[0];
    const int*   ei    = (const int*)d_in[1];   // [2,E] flat: src = ei[e], dst = ei[E+e]
    const int*   batch = (const int*)d_in[2];
    const float* W1    = (const float*)d_in[3];
    const float* b1    = (const float*)d_in[4];
    const float* W2    = (const float*)d_in[5];
    const float* b2    = (const float*)d_in[6];
    float* out = (float*)d_out;

    const int N = in_sizes[0] / 3;
    const int E = in_sizes[1] / 2;

    // workspace layout (floats), 64-float (256 B) aligned segments
    float* ws = (float*)d_ws;
    size_t Np  = ((size_t)N + 63) & ~(size_t)63;
    size_t ENp = ((size_t)(E + N) + 63) & ~(size_t)63;
    float* dinv   = ws;                 // [N]     deg -> dinv in place
    float* nrm    = dinv  + Np;         // [E+N]   per-edge norm cache
    float* agg3   = nrm   + ENp;        // [N,3]
    float* h1     = agg3  + ((Np * 3 + 63) & ~(size_t)63); // [N,64]
    float* agg64  = h1    + Np * 64;    // [N,64]
    float* pooled = agg64 + Np * 64;    // [128,64]
    float* counts = pooled + 128 * 64;  // [128]

    auto nb = [](long long n) { return (int)((n + BS - 1) / BS); };

    // init accumulators (fresh every call -> deterministic under graph replay)
    k_fill<<<nb(N), BS, 0, stream>>>(dinv, N, 1.0f);                 // self-loop degree
    k_fill<<<nb((long long)N * 3), BS, 0, stream>>>(agg3, N * 3, 0.f);
    k_fill<<<nb((long long)N * 64), BS, 0, stream>>>(agg64, N * 64, 0.f);
    k_fill<<<nb(128 * 64 + 128), BS, 0, stream>>>(pooled, 128 * 64 + 128, 0.f);

    // symmetric normalization
    k_degree<<<nb(E), BS, 0, stream>>>(ei, dinv, E);
    k_rsqrt<<<nb(N), BS, 0, stream>>>(dinv, N);

    // layer 1: aggregate narrow x, then WMMA linear + bias + relu
    k_edge3<<<nb((long long)(E + N)), BS, 0, stream>>>(ei, dinv, x, agg3, nrm, E, N);
    k_gemm1<<<(N + 31) / 32, BS, 0, stream>>>(agg3, W1, b1, h1, N);

    // layer 2: aggregate h1 (64-wide); GEMM deferred past pooling (linearity)
    k_edge64<<<nb((long long)(E + N) * 16), BS, 0, stream>>>(ei, nrm, h1, agg64, E, N);

    // pool -> mean -> tiny WMMA GEMM with fused bias writes d_out directly
    k_pool<<<nb((long long)N * 16), BS, 0, stream>>>(agg64, batch, pooled, counts, N);
    k_prep<<<nb(128 * 64), BS, 0, stream>>>(pooled, counts);
    k_gemm_out<<<8, BS, 0, stream>>>(pooled, W2, b2, counts, out);
}